// FMMLinear_54039278519042
// MI455X (gfx1250) — compile-verified
//
#include <hip/hip_runtime.h>

typedef float v2f __attribute__((ext_vector_type(2)));
typedef float v4f __attribute__((ext_vector_type(4)));
typedef float v8f __attribute__((ext_vector_type(8)));

#define IN_CH  4096
#define OUT_CH 4096
#define N_TOK  8192
#define RANK   16

// ---------------------------------------------------------------------------
// Pass 1: T[8192,16] = x @ right^T        (bandwidth: read x = 128 MB, NT)
// grid.x = 512 token tiles, 256 threads = 8 waves; wave w covers K in
// [w*512, (w+1)*512) in steps of 8 (two V_WMMA_F32_16X16X4_F32 per step).
//
// K-permutation trick: WMMA's K reduction is order-free, so lane L loads a
// contiguous float4 at (k + 4*hi) and feeds {x,y} to wmma0 and {z,w} to
// wmma1; A and B use identical addressing, so the K assignment matches.
//   A lane L: M = L&15, hi = L>>4 supplies logical K = {2hi, 2hi+1}
//   B lane L: N = L&15 (B[k][n] = right[n][k])
// 8 per-wave partial 16x16 tiles reduced deterministically via LDS.
// ---------------------------------------------------------------------------
__global__ __launch_bounds__(256) void lowrank_pass1(const float* __restrict__ x,
                                                     const float* __restrict__ right,
                                                     float* __restrict__ T)
{
    __shared__ float partial[8 * 256];   // 8 KB

    const int lane  = threadIdx.x & 31;
    const int wave  = threadIdx.x >> 5;
    const int ttile = blockIdx.x;
    const int m     = lane & 15;
    const int hi    = lane >> 4;

    const float* xrow = x     + (size_t)(ttile * 16 + m) * IN_CH + 4 * hi;
    const float* rrow = right + (size_t)m * IN_CH + 4 * hi;

    v8f c = {};
    const int kbase = wave * 512;
    #pragma unroll 4
    for (int it = 0; it < 64; ++it) {
        const int k = kbase + it * 8;
        // x: streamed once globally -> non-temporal; right: reused by all
        // 512 blocks (256 KB) -> default RT, stays L2-resident.
        v4f a4 = __builtin_nontemporal_load((const v4f*)(xrow + k));
        v4f b4 = *(const v4f*)(rrow + k);
        v2f a0 = __builtin_shufflevector(a4, a4, 0, 1);
        v2f a1 = __builtin_shufflevector(a4, a4, 2, 3);
        v2f b0 = __builtin_shufflevector(b4, b4, 0, 1);
        v2f b1 = __builtin_shufflevector(b4, b4, 2, 3);
        c = __builtin_amdgcn_wmma_f32_16x16x4_f32(false, a0, false, b0,
                                                  (short)0, c, false, false);
        c = __builtin_amdgcn_wmma_f32_16x16x4_f32(false, a1, false, b1,
                                                  (short)0, c, false, false);
    }

    // C/D layout: VGPR v, lanes 0-15 -> M=v ; lanes 16-31 -> M=v+8 ; N = lane&15
    #pragma unroll
    for (int v = 0; v < 8; ++v)
        partial[wave * 256 + (v + 8 * hi) * 16 + m] = c[v];
    __syncthreads();

    float s = 0.f;
    #pragma unroll
    for (int w = 0; w < 8; ++w) s += partial[w * 256 + threadIdx.x];
    const int em = threadIdx.x >> 4;     // M (token within tile)
    const int en = threadIdx.x & 15;     // N (rank)
    T[(size_t)(ttile * 16 + em) * RANK + en] = s;   // RT: pass 2 re-reads from L2
}

// ---------------------------------------------------------------------------
// Pass 2: out[8192,4096] = T @ left^T + bias   (bandwidth: write out = 128 MB, NT)
// One wave per 16x16 output tile; rank=16 -> 4 WMMAs fed by 2 b128 loads of T
// and 2 of left (same K-permutation trick). T (512 KB) and left (256 KB) are
// L2-resident; this pass is write-bound.
// ---------------------------------------------------------------------------
__global__ __launch_bounds__(256) void lowrank_pass2(const float* __restrict__ T,
                                                     const float* __restrict__ left,
                                                     const float* __restrict__ bias,
                                                     float* __restrict__ out)
{
    const int lane   = threadIdx.x & 31;
    const int wave   = threadIdx.x >> 5;
    const int tileId = blockIdx.x * 8 + wave;    // 131072 tiles total
    const int otile  = tileId & 255;             // OUT_CH/16 = 256
    const int ttile  = tileId >> 8;              // N_TOK/16  = 512
    const int m      = lane & 15;
    const int hi     = lane >> 4;

    const float* Trow = T    + (size_t)(ttile * 16 + m) * RANK + 4 * hi;
    const float* Lrow = left + (size_t)(otile * 16 + m) * RANK + 4 * hi;

    v8f c = {};
    #pragma unroll
    for (int kc = 0; kc < 2; ++kc) {             // two 8-wide K groups
        v4f a4 = *(const v4f*)(Trow + kc * 8);
        v4f b4 = *(const v4f*)(Lrow + kc * 8);
        v2f a0 = __builtin_shufflevector(a4, a4, 0, 1);
        v2f a1 = __builtin_shufflevector(a4, a4, 2, 3);
        v2f b0 = __builtin_shufflevector(b4, b4, 0, 1);
        v2f b1 = __builtin_shufflevector(b4, b4, 2, 3);
        c = __builtin_amdgcn_wmma_f32_16x16x4_f32(false, a0, false, b0,
                                                  (short)0, c, false, false);
        c = __builtin_amdgcn_wmma_f32_16x16x4_f32(false, a1, false, b1,
                                                  (short)0, c, false, false);
    }

    const float bv = bias[otile * 16 + m];       // N = lane&15, same for all 8 VGPRs
    float* obase = out + (size_t)(ttile * 16) * OUT_CH + otile * 16 + m;
    #pragma unroll
    for (int v = 0; v < 8; ++v)                  // write-once stream -> NT stores
        __builtin_nontemporal_store(c[v] + bv, obase + (size_t)(v + 8 * hi) * OUT_CH);
}

// ---------------------------------------------------------------------------
extern "C" void kernel_launch(void* const* d_in, const int* in_sizes, int n_in,
                              void* d_out, int out_size, void* d_ws, size_t ws_size,
                              hipStream_t stream)
{
    const float* x     = (const float*)d_in[0];   // [8192, 4096]
    const float* left  = (const float*)d_in[1];   // [4096, 16]
    const float* right = (const float*)d_in[2];   // [16, 4096]
    const float* bias  = (const float*)d_in[3];   // [4096]
    float*       out   = (float*)d_out;           // [8192, 4096]
    float*       T     = (float*)d_ws;            // [8192, 16] = 512 KB scratch

    lowrank_pass1<<<N_TOK / 16, 256, 0, stream>>>(x, right, T);

    const int tiles = (N_TOK / 16) * (OUT_CH / 16);   // 131072
    lowrank_pass2<<<tiles / 8, 256, 0, stream>>>(T, left, bias, out);
}